// MACETensorInteraction_16819091931701
// MI455X (gfx1250) — compile-verified
//
#include <hip/hip_runtime.h>
#include <hip/hip_bf16.h>

typedef __attribute__((ext_vector_type(2))) float v2f;
typedef __attribute__((ext_vector_type(4))) float v4f;
typedef __attribute__((ext_vector_type(8))) float v8f;

#define N_NODES 20000
#define N_EDGES 200000
#define NTILES  ((N_EDGES + 15) / 16)          // 12500
#define WAVES_PER_WG 4
#define TILES_PER_WAVE 2
#define TILES_PER_WG (WAVES_PER_WG * TILES_PER_WAVE) // 8
#define KSTEPS 17                               // 68 hidden channels / 4 (64 + bias row + zero pad)

#define INV_SQRT3 0.57735026918962576f
#define PW0 0.17677669529663687f   // sqrt(1/32)
#define PW1 0.30618621784789724f   // sqrt(3/32)
#define CV  (PW1 * INV_SQRT3)
#define SI_NORM 0.25f              // 1/sqrt(16)

// ---------------------------------------------------------------------------
// Kernel 0: pre-swizzle w2 (64x1024) + b2 (1024) into WMMA B-fragment layout.
// Layout: ws[((blk*17 + s)*32 + lane)*2 + {0,1}]
//   blk = k*16+u (column block of 16), s = K-step (4 channels), lane = wave lane.
//   c0 = 4*s + (lane>>4)*2 ; value0 = row c0, value1 = row c0+1, col = blk*16+(lane&15)
//   rows 0..63 = w2, row 64 = b2 (bias, pairs with h==1), rows 65..67 = 0.
// ---------------------------------------------------------------------------
__global__ __launch_bounds__(256) void swz_kernel(const float* __restrict__ w2,
                                                  const float* __restrict__ b2,
                                                  float* __restrict__ ws) {
  int i = blockIdx.x * 256 + threadIdx.x;
  if (i >= 64 * KSTEPS * 32) return;
  int lane = i & 31;
  int s    = (i >> 5) % KSTEPS;
  int blk  = i / (KSTEPS * 32);
  int c0   = 4 * s + ((lane >> 4) << 1);
  int j    = blk * 16 + (lane & 15);
  float v0 = (c0 < 64) ? w2[c0 * 1024 + j] : ((c0 == 64) ? b2[j] : 0.f);
  int c1 = c0 + 1;
  float v1 = (c1 < 64) ? w2[c1 * 1024 + j] : ((c1 == 64) ? b2[j] : 0.f);
  ws[(size_t)i * 2 + 0] = v0;
  ws[(size_t)i * 2 + 1] = v1;
}

// ---------------------------------------------------------------------------
// Kernel 1: self-interaction, writes (not adds) the output base values.
// ---------------------------------------------------------------------------
__global__ __launch_bounds__(256) void si_kernel(const float* __restrict__ nf,
                                                 const float* __restrict__ wsi0,
                                                 const float* __restrict__ wsi1,
                                                 float* __restrict__ out) {
  int i = blockIdx.x * 256 + threadIdx.x;
  if (i >= N_NODES * 64) return;
  int n = i >> 6, c = i & 63;
  const float* x = nf + (size_t)n * 64;
  float s = 0.f;
  if (c < 16) {
#pragma unroll
    for (int u = 0; u < 16; ++u) s += x[u] * wsi0[u * 16 + c];
  } else {
    int cc = c - 16, w = cc / 3, m = cc % 3;
#pragma unroll
    for (int u = 0; u < 16; ++u) s += x[16 + u * 3 + m] * wsi1[u * 16 + w];
  }
  out[i] = s * SI_NORM;
}

// ---------------------------------------------------------------------------
// Kernel 2: fused edge kernel. One wave owns 2 tiles of 16 edges.
// Per wave-private LDS region (1536 floats per tile):
//   phase 1: h staging [e][68] (1088 floats) -> A-fragments (transpose)
//   phase 2 (overwrites): a-vectors [type 0..5][u 0..15][e 0..15]
// types: 0=xs*y0  1=xs  2..4=xv_m  5=INV_SQRT3*dot
// ---------------------------------------------------------------------------
__device__ __forceinline__ void fma8(v8f& acc, const float* areg, int type, int u,
                                     int uh, const v8f& c) {
  const v4f* ap = (const v4f*)(areg + (type * 16 + u) * 16 + uh * 8);
  v4f lo = ap[0], hi = ap[1];
  acc[0] += lo[0] * c[0]; acc[1] += lo[1] * c[1];
  acc[2] += lo[2] * c[2]; acc[3] += lo[3] * c[3];
  acc[4] += hi[0] * c[4]; acc[5] += hi[1] * c[5];
  acc[6] += hi[2] * c[6]; acc[7] += hi[3] * c[7];
}

__global__ __launch_bounds__(WAVES_PER_WG * 32) void edge_kernel(
    const float* __restrict__ nf, const float* __restrict__ ea,
    const float* __restrict__ sh, const float* __restrict__ w1,
    const float* __restrict__ b1, const int* __restrict__ rowi,
    const int* __restrict__ coli, const float* __restrict__ bsw,
    float* __restrict__ out) {
  __shared__ float smem[TILES_PER_WG * 1536];
  const int lane = threadIdx.x & 31;
  const int wave = threadIdx.x >> 5;
  const int uh = lane >> 4;   // lane half
  const int lm = lane & 15;
  const int tile0 = (blockIdx.x * WAVES_PER_WG + wave) * TILES_PER_WAVE;

  float* aw[TILES_PER_WAVE];
#pragma unroll
  for (int t = 0; t < TILES_PER_WAVE; ++t)
    aw[t] = smem + (wave * TILES_PER_WAVE + t) * 1536;

  // hoisted MLP weights: lane owns channel c = ch*32 + lane
  float w1r[2][8], b1r[2];
#pragma unroll
  for (int ch = 0; ch < 2; ++ch) {
    b1r[ch] = b1[ch * 32 + lane];
#pragma unroll
    for (int b = 0; b < 8; ++b) w1r[ch][b] = w1[b * 64 + ch * 32 + lane];
  }

  v2f afr[TILES_PER_WAVE][KSTEPS];

#pragma unroll
  for (int t = 0; t < TILES_PER_WAVE; ++t) {
    const int e0 = (tile0 + t) * 16;
    float* hreg = aw[t];  // h staging aliases the a-region (dead before a stores)

    // ---- MLP: h[e][c] = silu(ea[e] . w1[:,c] + b1[c]) ----
    for (int e = 0; e < 16; ++e) {
      int ee = e0 + e; if (ee >= N_EDGES) ee = N_EDGES - 1;
      v4f ea0 = *(const v4f*)(ea + (size_t)ee * 8);
      v4f ea1 = *(const v4f*)(ea + (size_t)ee * 8 + 4);
      float eav[8] = {ea0[0], ea0[1], ea0[2], ea0[3], ea1[0], ea1[1], ea1[2], ea1[3]};
#pragma unroll
      for (int ch = 0; ch < 2; ++ch) {
        float acc = b1r[ch];
#pragma unroll
        for (int b = 0; b < 8; ++b) acc += eav[b] * w1r[ch][b];
        float hv = acc / (1.f + __expf(-acc));  // silu
        hreg[e * 68 + ch * 32 + lane] = hv;
      }
    }
    // bias column: h[e][64]=1, 65..67 = 0
    if (lane < 16) *(v4f*)(hreg + lane * 68 + 64) = (v4f){1.f, 0.f, 0.f, 0.f};

    // ---- transpose into A fragments (16x4 per K-step) ----
    // A layout: VGPR0 = h[m][4s + uh*2], VGPR1 = +1  (m = lane&15)
#pragma unroll
    for (int s = 0; s < KSTEPS; ++s)
      afr[t][s] = *(const v2f*)(hreg + lm * 68 + s * 4 + uh * 2);

    // ---- build a-vectors (lane: edge=lm, u range uh*8..uh*8+7) ----
    {
      int ge = e0 + lm;
      bool valid = ge < N_EDGES;
      int gec = valid ? ge : 0;
      int cidx = coli[gec];
      const float* xr = nf + (size_t)cidx * 64;
      v4f xa = *(const v4f*)(xr + uh * 8);
      v4f xb = *(const v4f*)(xr + uh * 8 + 4);
      v4f xv[6];
#pragma unroll
      for (int i = 0; i < 6; ++i)
        xv[i] = *(const v4f*)(xr + 16 + uh * 24 + 4 * i);
      v4f yv = *(const v4f*)(sh + (size_t)gec * 4);
      if (!valid) {
        xa = (v4f)0.f; xb = (v4f)0.f; yv = (v4f)0.f;
#pragma unroll
        for (int i = 0; i < 6; ++i) xv[i] = (v4f)0.f;
      }
      float xs[8] = {xa[0], xa[1], xa[2], xa[3], xb[0], xb[1], xb[2], xb[3]};
      float xvf[24];
#pragma unroll
      for (int i = 0; i < 6; ++i) {
        xvf[4 * i + 0] = xv[i][0]; xvf[4 * i + 1] = xv[i][1];
        xvf[4 * i + 2] = xv[i][2]; xvf[4 * i + 3] = xv[i][3];
      }
      float y0 = yv[0];
      float* A = aw[t];
#pragma unroll
      for (int ul = 0; ul < 8; ++ul) {
        int u = uh * 8 + ul;
        float d = xvf[ul * 3] * yv[1] + xvf[ul * 3 + 1] * yv[2] + xvf[ul * 3 + 2] * yv[3];
        A[(0 * 16 + u) * 16 + lm] = xs[ul] * y0;
        A[(1 * 16 + u) * 16 + lm] = xs[ul];
        A[(2 * 16 + u) * 16 + lm] = xvf[ul * 3 + 0];
        A[(3 * 16 + u) * 16 + lm] = xvf[ul * 3 + 1];
        A[(4 * 16 + u) * 16 + lm] = xvf[ul * 3 + 2];
        A[(5 * 16 + u) * 16 + lm] = INV_SQRT3 * d;
      }
    }
  }

  // ---- main loop: 64 (k,u) blocks, GEMM tile + on-the-fly contraction ----
  v8f acc0[2] = {(v8f)0.f, (v8f)0.f};
  v8f acc1[2] = {(v8f)0.f, (v8f)0.f};
  v8f acc2[2][3];
#pragma unroll
  for (int t = 0; t < 2; ++t)
#pragma unroll
    for (int m = 0; m < 3; ++m) acc2[t][m] = (v8f)0.f;

  const v2f* bp = (const v2f*)bsw;
#pragma unroll
  for (int k = 0; k < 4; ++k) {
#pragma unroll 1
    for (int u = 0; u < 16; ++u) {
      const int blk = k * 16 + u;
      v2f bf[KSTEPS];
#pragma unroll
      for (int s = 0; s < KSTEPS; ++s)
        bf[s] = bp[(size_t)(blk * KSTEPS + s) * 32 + lane];
      v8f cA = (v8f)0.f, cB = (v8f)0.f;
#pragma unroll
      for (int s = 0; s < KSTEPS; ++s) {
        cA = __builtin_amdgcn_wmma_f32_16x16x4_f32(false, afr[0][s], false, bf[s],
                                                   (short)0, cA, false, false);
        cB = __builtin_amdgcn_wmma_f32_16x16x4_f32(false, afr[1][s], false, bf[s],
                                                   (short)0, cB, false, false);
      }
      if (k == 0) {
        fma8(acc0[0], aw[0], 0, u, uh, cA);
        fma8(acc0[1], aw[1], 0, u, uh, cB);
      } else if (k == 1) {
        fma8(acc1[0], aw[0], 1, u, uh, cA);
        fma8(acc1[1], aw[1], 1, u, uh, cB);
      } else if (k == 2) {
#pragma unroll
        for (int m = 0; m < 3; ++m) {
          fma8(acc2[0][m], aw[0], 2 + m, u, uh, cA);
          fma8(acc2[1][m], aw[1], 2 + m, u, uh, cB);
        }
      } else {
        fma8(acc0[0], aw[0], 5, u, uh, cA);
        fma8(acc0[1], aw[1], 5, u, uh, cB);
      }
    }
  }

  // ---- epilogue: scale, combine with y0/y1, scatter-add ----
#pragma unroll
  for (int t = 0; t < 2; ++t) {
    const int e0 = (tile0 + t) * 16;
#pragma unroll
    for (int r = 0; r < 8; ++r) {
      int ge = e0 + r + uh * 8;
      bool valid = ge < N_EDGES;
      int gec = valid ? ge : 0;
      v4f yv = *(const v4f*)(sh + (size_t)gec * 4);
      int node = rowi[gec];
      float* op = out + (size_t)node * 64;
      float o0 = valid ? PW0 * acc0[t][r] : 0.f;
      unsafeAtomicAdd(op + lm, o0);
#pragma unroll
      for (int m = 0; m < 3; ++m) {
        float ov = CV * (acc1[t][r] * yv[1 + m] + acc2[t][m][r] * yv[0]);
        unsafeAtomicAdd(op + 16 + lm * 3 + m, valid ? ov : 0.f);
      }
    }
  }
}

// ---------------------------------------------------------------------------
extern "C" void kernel_launch(void* const* d_in, const int* in_sizes, int n_in,
                              void* d_out, int out_size, void* d_ws, size_t ws_size,
                              hipStream_t stream) {
  const float* node_feat = (const float*)d_in[0];
  const float* edge_attr = (const float*)d_in[1];
  const float* sh        = (const float*)d_in[2];
  const float* w1        = (const float*)d_in[3];
  const float* b1        = (const float*)d_in[4];
  const float* w2        = (const float*)d_in[5];
  const float* b2        = (const float*)d_in[6];
  const float* wsi0      = (const float*)d_in[7];
  const float* wsi1      = (const float*)d_in[8];
  const int*   eidx      = (const int*)d_in[9];
  const int* row = eidx;
  const int* col = eidx + N_EDGES;
  float* bsw = (float*)d_ws;   // 64*17*32*2 floats = 278528 B
  float* out = (float*)d_out;

  si_kernel<<<(N_NODES * 64 + 255) / 256, 256, 0, stream>>>(node_feat, wsi0, wsi1, out);
  swz_kernel<<<(64 * KSTEPS * 32 + 255) / 256, 256, 0, stream>>>(w2, b2, bsw);

  int nwg = (NTILES + TILES_PER_WG - 1) / TILES_PER_WG;  // 1563
  edge_kernel<<<nwg, WAVES_PER_WG * 32, 0, stream>>>(
      node_feat, edge_attr, sh, w1, b1, row, col, bsw, out);
}